// gemGAT_Corrected_69767448756235
// MI455X (gfx1250) — compile-verified
//
#include <hip/hip_runtime.h>
#include <hip/hip_bf16.h>
#include <math.h>

typedef __attribute__((ext_vector_type(2))) float v2f;
typedef __attribute__((ext_vector_type(8))) float v8f;

#define NN   12000
#define EE   192000
#define NGI_ 4000
#define NGO_ 8000
#define DHEAD 128

// ---------- device helpers ----------
__device__ __forceinline__ float elu_f(float x) { return x > 0.f ? x : (__expf(x) - 1.f); }

__device__ __forceinline__ unsigned f2ord(float f) {
  unsigned u = __float_as_uint(f);
  return (u & 0x80000000u) ? ~u : (u | 0x80000000u);
}
__device__ __forceinline__ float ord2f(unsigned u) {
  unsigned v = (u & 0x80000000u) ? (u & 0x7fffffffu) : ~u;
  return __uint_as_float(v);
}
__device__ __forceinline__ float act_apply(float x, int act) {
  return (act == 1) ? elu_f(x) : x;
}
__device__ __forceinline__ float sigmoid_f(float x) { return 1.f / (1.f + __expf(-x)); }

#define WMMA_F32(A_, B_, C_) \
  __builtin_amdgcn_wmma_f32_16x16x4_f32(false, (A_), false, (B_), (short)0, (C_), false, false)

// ---------- fast WMMA GEMM: C[M,F] = A[M,K] @ W[K,F] (+bias)(+ELU) ----------
// Preconditions: M%32==0, K%4==0, F%16==0.
// One wave computes a 32x16 tile (two row tiles share one B fragment, so the
// strided W loads are amortized 2x). 8 waves/block tile along F.
__global__ void k_gemm_wmma_fast(const float* __restrict__ A, const float* __restrict__ W,
                                 const float* __restrict__ bias, float* __restrict__ C,
                                 int M, int K, int F, int act, int has_bias)
{
  const int lane = threadIdx.x & 31;
  const int wave = threadIdx.x >> 5;
  const int r    = lane & 15;
  const int half = lane >> 4;
  const int tile_f = blockIdx.x * 8 + wave;
  const int col0   = tile_f * 16;
  if (col0 >= F) return;                        // uniform per-wave exit
  const int row0 = blockIdx.y * 32;

  const float* Ap0 = A + (long long)(row0 + r) * K + 2 * half;  // rows row0..row0+15
  const float* Ap1 = Ap0 + (long long)16 * K;                   // rows row0+16..row0+31
  const float* wp  = W + (long long)(2 * half) * F + col0 + r;  // strength-reduced W ptr
  const long long wstep = 4LL * F;

  v8f acc0 = {}, acc1 = {};
#pragma unroll 4
  for (int k0 = 0; k0 < K; k0 += 4) {
    const v2f a0 = *(const v2f*)(Ap0 + k0);
    const v2f a1 = *(const v2f*)(Ap1 + k0);
    v2f b;
    b.x = wp[0];
    b.y = wp[F];
    wp += wstep;
    acc0 = WMMA_F32(a0, b, acc0);
    acc1 = WMMA_F32(a1, b, acc1);
  }

  const float bb = has_bias ? bias[col0 + r] : 0.f;
  float* Cp = C + (long long)row0 * F + col0 + r;
#pragma unroll
  for (int v = 0; v < 8; ++v) {
    const long long roff = (long long)(v + 8 * half) * F;
    Cp[roff]                  = act_apply(acc0[v] + bb, act);
    Cp[roff + 16LL * F]       = act_apply(acc1[v] + bb, act);
  }
}

// ---------- outer product (K==1): C[M,F] = A[M,1]*W[1,F] (+bias)(+act) ----------
__global__ void k_outer1(const float* __restrict__ A, const float* __restrict__ W,
                         const float* __restrict__ bias, float* __restrict__ C,
                         int M, int F, int act, int has_bias)
{
  const long long g = (long long)blockIdx.x * blockDim.x + threadIdx.x;
  if (g >= (long long)M * F) return;
  const int f = (int)(g % F);
  const int m = (int)(g / F);
  float x = A[m] * W[f];
  if (has_bias) x += bias[f];
  C[g] = act_apply(x, act);
}

// ---------- small-F naive GEMM (F < 16; tiny tail layers) ----------
__global__ void k_gemm_small(const float* __restrict__ A, const float* __restrict__ W,
                             const float* __restrict__ bias, float* __restrict__ C,
                             int M, int K, int F, int act, int has_bias)
{
  const long long g = (long long)blockIdx.x * blockDim.x + threadIdx.x;
  if (g >= (long long)M * F) return;
  const int f = (int)(g % F);
  const int m = (int)(g / F);
  const float* ar = A + (long long)m * K;
  float s = has_bias ? bias[f] : 0.f;
  for (int k = 0; k < K; ++k) s += ar[k] * W[k * F + f];
  C[g] = act_apply(s, act);
}

// ---------- WMMA NT GEMM: C[M,NC] = sigmoid(A[M,128] @ B[NC,128]^T) ----------
// One wave computes a 32x32 macro-tile (2x2 16x16 tiles, 4 accumulators).
// Per k-step: 4 contiguous b64 loads feed 4 WMMAs. Preconditions: M%32==0, NC%32==0.
__global__ void k_gemm_nt_sig(const float* __restrict__ A, const float* __restrict__ B,
                              float* __restrict__ C, int M, int NC)
{
  const int lane = threadIdx.x & 31;
  const int wave = threadIdx.x >> 5;
  const int r    = lane & 15;
  const int half = lane >> 4;
  const int pair = blockIdx.x * 8 + wave;
  const int col0 = pair * 32;
  if (col0 >= NC) return;
  const int row0 = blockIdx.y * 32;

  const float* Ap0 = A + (long long)(row0 + r) * DHEAD + 2 * half;
  const float* Ap1 = Ap0 + 16LL * DHEAD;
  const float* Bp0 = B + (long long)(col0 + r) * DHEAD + 2 * half;
  const float* Bp1 = Bp0 + 16LL * DHEAD;

  v8f acc00 = {}, acc01 = {}, acc10 = {}, acc11 = {};
#pragma unroll 8
  for (int k0 = 0; k0 < DHEAD; k0 += 4) {
    const v2f a0 = *(const v2f*)(Ap0 + k0);
    const v2f a1 = *(const v2f*)(Ap1 + k0);
    const v2f b0 = *(const v2f*)(Bp0 + k0);
    const v2f b1 = *(const v2f*)(Bp1 + k0);
    acc00 = WMMA_F32(a0, b0, acc00);
    acc01 = WMMA_F32(a0, b1, acc01);
    acc10 = WMMA_F32(a1, b0, acc10);
    acc11 = WMMA_F32(a1, b1, acc11);
  }

  float* Cp = C + (long long)row0 * NC + col0 + r;
#pragma unroll
  for (int v = 0; v < 8; ++v) {
    const long long roff = (long long)(v + 8 * half) * NC;
    Cp[roff]                  = sigmoid_f(acc00[v]);
    Cp[roff + 16]             = sigmoid_f(acc01[v]);
    Cp[roff + 16LL * NC]      = sigmoid_f(acc10[v]);
    Cp[roff + 16LL * NC + 16] = sigmoid_f(acc11[v]);
  }
}

// ---------- GAT attention kernels ----------
__global__ void k_el_er(const float* __restrict__ H, const float* __restrict__ al,
                        const float* __restrict__ ar, float* __restrict__ el,
                        float* __restrict__ er, int heads)
{
  const int g = blockIdx.x * blockDim.x + threadIdx.x;  // n*heads + hd
  if (g >= NN * heads) return;
  const int hd = g % heads;
  const int n  = g / heads;
  const float* hrow = H + ((long long)n * heads + hd) * DHEAD;
  const float* pal  = al + hd * DHEAD;
  const float* par  = ar + hd * DHEAD;
  float sl = 0.f, sr = 0.f;
  for (int j = 0; j < DHEAD; ++j) {
    const float v = hrow[j];
    sl += v * pal[j];
    sr += v * par[j];
  }
  el[g] = sl;
  er[g] = sr;
}

__global__ void k_seg_init(unsigned* __restrict__ mord, float* __restrict__ s, int n)
{
  const int g = blockIdx.x * blockDim.x + threadIdx.x;
  if (g < n) { mord[g] = 0x007FFFFFu; /* ord(-inf) */ s[g] = 0.f; }
}

__global__ void k_edge_logits(const float* __restrict__ el, const float* __restrict__ er,
                              const int* __restrict__ src, const int* __restrict__ dst,
                              float* __restrict__ ebuf, unsigned* __restrict__ mord, int heads)
{
  const int g = blockIdx.x * blockDim.x + threadIdx.x;
  if (g >= EE * heads) return;
  const int hd = g % heads;
  const int e  = g / heads;
  const int sn = src[e], dn = dst[e];
  float x = el[sn * heads + hd] + er[dn * heads + hd];
  x = (x >= 0.f) ? x : 0.2f * x;                     // leaky_relu 0.2
  ebuf[g] = x;
  atomicMax(&mord[dn * heads + hd], f2ord(x));
}

__global__ void k_decode_m(unsigned* __restrict__ mord, int n)
{
  const int g = blockIdx.x * blockDim.x + threadIdx.x;
  if (g >= n) return;
  float f = ord2f(mord[g]);
  if (!isfinite(f)) f = 0.f;                          // reference: where(isfinite, m, 0)
  ((float*)mord)[g] = f;
}

__global__ void k_edge_w(float* __restrict__ ebuf, const float* __restrict__ m,
                         const int* __restrict__ dst, float* __restrict__ s, int heads)
{
  const int g = blockIdx.x * blockDim.x + threadIdx.x;
  if (g >= EE * heads) return;
  const int hd = g % heads;
  const int e  = g / heads;
  const int dn = dst[e];
  const float w = __expf(ebuf[g] - m[dn * heads + hd]);
  ebuf[g] = w;
  atomicAdd(&s[dn * heads + hd], w);
}

__global__ void k_zero(float* __restrict__ p, long long n)
{
  const long long g = (long long)blockIdx.x * blockDim.x + threadIdx.x;
  if (g < n) p[g] = 0.f;
}

__global__ void k_edge_scatter(const float* __restrict__ H, const float* __restrict__ wbuf,
                               const float* __restrict__ s, const int* __restrict__ src,
                               const int* __restrict__ dst, float* __restrict__ out, int heads)
{
  const long long g = (long long)blockIdx.x * blockDim.x + threadIdx.x;
  const long long total = (long long)EE * heads * DHEAD;
  if (g >= total) return;
  const int j = (int)(g & (DHEAD - 1));
  const long long t = g >> 7;                         // e*heads + hd
  const int hd = (int)(t % heads);
  const long long e = t / heads;
  const int sn = src[e], dn = dst[e];
  const float alpha = wbuf[t] / fmaxf(s[dn * heads + hd], 1e-16f);
  atomicAdd(&out[((long long)dn * heads + hd) * DHEAD + j],
            H[((long long)sn * heads + hd) * DHEAD + j] * alpha);
}

__global__ void k_bias_act(float* __restrict__ out, const float* __restrict__ b, int F, int act)
{
  const long long g = (long long)blockIdx.x * blockDim.x + threadIdx.x;
  if (g >= (long long)NN * F) return;
  const int f = (int)(g % F);
  out[g] = act_apply(out[g] + b[f], act);
}

__global__ void k_imputed(const float* __restrict__ pred, const float* __restrict__ x,
                          float* __restrict__ imp)
{
  const int g = blockIdx.x * blockDim.x + threadIdx.x;
  if (g >= NN) return;
  imp[g] = (g < NGI_) ? pred[g] : x[g];
}

// ---------- host-side helpers ----------
static inline int cdiv(long long a, long long b) { return (int)((a + b - 1) / b); }

static void launch_gemm(const float* A, const float* W, const float* bias, float* C,
                        int M, int K, int F, int act, hipStream_t st)
{
  const int has_bias = (bias != nullptr);
  if (K == 1) {
    k_outer1<<<cdiv((long long)M * F, 256), 256, 0, st>>>(A, W, bias, C, M, F, act, has_bias);
  } else if (F < 16) {
    k_gemm_small<<<cdiv((long long)M * F, 256), 256, 0, st>>>(A, W, bias, C, M, K, F, act,
                                                              has_bias);
  } else {
    dim3 grid(cdiv(F / 16, 8), M / 32);
    k_gemm_wmma_fast<<<grid, 256, 0, st>>>(A, W, bias, C, M, K, F, act, has_bias);
  }
}

static void launch_nt(const float* A, const float* B, float* C, int M, int NC, hipStream_t st)
{
  dim3 grid(cdiv(NC / 32, 8), M / 32);
  k_gemm_nt_sig<<<grid, 256, 0, st>>>(A, B, C, M, NC);
}

struct GP { const float *W, *al, *ar, *b; };
struct Small { float *el, *er, *m, *s, *ebuf; };

static void run_gat(const float* X, int Fin, const GP& p, int heads,
                    const int* src, const int* dst,
                    float* H, float* OUT, int elu, const Small& sb, hipStream_t st)
{
  const int F  = heads * DHEAD;
  const int nh = NN * heads;
  const int eh = EE * heads;
  launch_gemm(X, p.W, nullptr, H, NN, Fin, F, 0, st);
  k_el_er<<<cdiv(nh, 256), 256, 0, st>>>(H, p.al, p.ar, sb.el, sb.er, heads);
  k_seg_init<<<cdiv(nh, 256), 256, 0, st>>>((unsigned*)sb.m, sb.s, nh);
  k_edge_logits<<<cdiv(eh, 256), 256, 0, st>>>(sb.el, sb.er, src, dst, sb.ebuf,
                                               (unsigned*)sb.m, heads);
  k_decode_m<<<cdiv(nh, 256), 256, 0, st>>>((unsigned*)sb.m, nh);
  k_edge_w<<<cdiv(eh, 256), 256, 0, st>>>(sb.ebuf, sb.m, dst, sb.s, heads);
  const long long tot = (long long)NN * F;
  k_zero<<<cdiv(tot, 256), 256, 0, st>>>(OUT, tot);
  const long long scat = (long long)eh * DHEAD;
  k_edge_scatter<<<cdiv(scat, 256), 256, 0, st>>>(H, sb.ebuf, sb.s, src, dst, OUT, heads);
  k_bias_act<<<cdiv(tot, 256), 256, 0, st>>>(OUT, p.b, F, elu);
}

extern "C" void kernel_launch(void* const* d_in, const int* in_sizes, int n_in,
                              void* d_out, int out_size, void* d_ws, size_t ws_size,
                              hipStream_t stream)
{
  const float* x   = (const float*)d_in[0];
  const int*   eis = (const int*)d_in[1];
  const int*   eic = (const int*)d_in[2];
  const int *srcS = eis, *dstS = eis + EE;
  const int *srcC = eic, *dstC = eic + EE;

  int i = 3;
  auto getG = [&]() {
    GP g;
    g.W  = (const float*)d_in[i++];
    g.al = (const float*)d_in[i++];
    g.ar = (const float*)d_in[i++];
    g.b  = (const float*)d_in[i++];
    return g;
  };
  const GP enc1 = getG(), enc2 = getG(), enc3 = getG(), enc4 = getG();
  const GP link1 = getG(), link2 = getG(), semi1 = getG(), semi2 = getG();
  const float* predin[16];   for (int k = 0; k < 16; ++k) predin[k]   = (const float*)d_in[i++];
  const float* predlink[6];  for (int k = 0; k < 6;  ++k) predlink[k] = (const float*)d_in[i++];
  const float* predout[16];  for (int k = 0; k < 16; ++k) predout[k]  = (const float*)d_in[i++];

  // workspace layout (floats)
  float* ws = (float*)d_ws;
  const size_t big = (size_t)NN * 512;
  float* B0 = ws;
  float* B1 = B0 + big;
  float* B2 = B1 + big;
  Small sb;
  sb.el   = B2 + big;
  sb.er   = sb.el + (size_t)NN * 4;
  sb.m    = sb.er + (size_t)NN * 4;
  sb.s    = sb.m  + (size_t)NN * 4;
  sb.ebuf = sb.s  + (size_t)NN * 4;
  float* imput = sb.ebuf + (size_t)EE * 4;

  // output slices (concatenated in return order)
  float* out_pred_in  = (float*)d_out;                       // 4000
  float* out_pred_all = (float*)d_out + 4000;                // 12000
  float* out_ss = (float*)d_out + 16000;                     // 8000*8000
  float* out_st = out_ss + (size_t)NGO_ * NGO_;              // 8000*4000
  float* out_tt = out_st + (size_t)NGO_ * (NN - NGO_);       // 4000*4000

  // ---- encoder on source graph ----
  run_gat(x,  1,   enc1, 4, srcS, dstS, B0, B1, 1, sb, stream);   // B1: N x 512
  run_gat(B1, 512, enc2, 1, srcS, dstS, B0, B2, 1, sb, stream);   // B2: N x 128
  run_gat(B2, 128, enc3, 4, srcS, dstS, B0, B1, 1, sb, stream);   // B1: N x 512
  run_gat(B1, 512, enc4, 1, srcS, dstS, B0, B2, 1, sb, stream);   // z = B2: N x 128

  // ---- pred_in MLP on z[:4000] ----
  {
    const int dims[9] = {128, 512, 256, 256, 256, 64, 16, 4, 1};
    const float* in = B2;
    float* pp[2] = {B0, B1};
    for (int l = 0; l < 8; ++l) {
      float* ob = (l == 7) ? out_pred_in : pp[l & 1];
      launch_gemm(in, predin[2 * l], predin[2 * l + 1], ob, NGI_, dims[l], dims[l + 1],
                  (l < 7) ? 1 : 0, stream);
      in = ob;
    }
  }
  k_imputed<<<cdiv(NN, 256), 256, 0, stream>>>(out_pred_in, x, imput);

  // ---- link branch on combined graph ----
  run_gat(imput, 1,   link1, 4, srcC, dstC, B0, B1, 1, sb, stream);  // B1: N x 512
  run_gat(B1,    512, link2, 1, srcC, dstC, B0, B2, 0, sb, stream);  // B2: N x 128 (no elu)
  launch_gemm(B2, predlink[0], predlink[1], B0, NN, 128, 128, 1, stream);
  launch_gemm(B0, predlink[2], predlink[3], B1, NN, 128, 128, 1, stream);
  launch_gemm(B1, predlink[4], predlink[5], B0, NN, 128, 128, 0, stream); // emb = B0

  // ---- adjacency products (WMMA f32, sigmoid epilogue, direct to d_out) ----
  const float* emb_s = B0;
  const float* emb_t = B0 + (size_t)NGO_ * 128;
  launch_nt(emb_s, emb_s, out_ss, NGO_, NGO_, stream);
  launch_nt(emb_s, emb_t, out_st, NGO_, NN - NGO_, stream);
  launch_nt(emb_t, emb_t, out_tt, NN - NGO_, NN - NGO_, stream);

  // ---- semi branch on combined graph ----
  run_gat(imput, 1,   semi1, 4, srcC, dstC, B1, B2, 1, sb, stream);  // B2: N x 512
  run_gat(B2,    512, semi2, 1, srcC, dstC, B1, B0, 0, sb, stream);  // s2 = B0: N x 128

  // ---- pred_out MLP on s2 (all N nodes) ----
  {
    const int dims[9] = {128, 512, 256, 256, 256, 64, 16, 4, 1};
    const float* in = B0;
    float* pp[2] = {B1, B2};
    for (int l = 0; l < 8; ++l) {
      float* ob = (l == 7) ? out_pred_all : pp[l & 1];
      launch_gemm(in, predout[2 * l], predout[2 * l + 1], ob, NN, dims[l], dims[l + 1],
                  (l < 7) ? 1 : 0, stream);
      in = ob;
    }
  }
}